// FrechetMean_7576322310235
// MI455X (gfx1250) — compile-verified
//
#include <hip/hip_runtime.h>

// Fréchet mean on the sphere, B=256 batches, N=1024 points, D=256 dims.
// One workgroup (256 threads = 8 wave32) per batch. Per iteration, points are
// streamed through LDS exactly once via the Tensor Data Mover (double-buffered,
// TENSORcnt-tracked); cos GEMV and grad GEMV are both computed from the LDS
// tile with V_WMMA_F32_16X16X4_F32.
// grad = sum_n a_n q_n - (sum_n a_n cos_n) * mean,  a_n = w_n * theta_n / ||u_n||.

typedef float v2f __attribute__((ext_vector_type(2)));
typedef float v8f __attribute__((ext_vector_type(8)));
typedef unsigned int v4u __attribute__((ext_vector_type(4)));
typedef int v8i __attribute__((ext_vector_type(8)));
typedef int v4i __attribute__((ext_vector_type(4)));

#define B_      256
#define N_      1024
#define D_      256
#define ITERS   10
#define LR      0.5f
#define EPSX    1e-7f
#define TILE_N  64
#define STRIDE  268            // 256 + 12 pad -> LDS bank-conflict-free fragments
#define NTILES  (N_ / TILE_N)

#if defined(__has_builtin)
#if __has_builtin(__builtin_amdgcn_tensor_load_to_lds)
#define USE_TDM 1
#endif
#endif
#ifndef USE_TDM
#define USE_TDM 0
#endif

#if USE_TDM
// Issue one TDM 2D tile load: 64 rows x 256 f32 from global into LDS, with the
// TDM pad unit inserting 12 DWORDs after every 256-DWORD row (-> stride 268).
__device__ __forceinline__ void tdm_load_tile(const float* gsrc, unsigned lds_byte_off) {
    unsigned long long ga = (unsigned long long)gsrc;
    v4u g0;
    g0.x = 1u;                                                 // count=1, no gather
    g0.y = lds_byte_off;                                       // lds_addr[63:32]
    g0.z = (unsigned)(ga & 0xFFFFFFFFull);                     // global_addr lo
    g0.w = (unsigned)((ga >> 32) & 0x1FFFFFFull) | (2u << 30); // addr hi | type=2
    v8i g1;
    g1[0] = (int)((2u << 16)        // data_size = 4 bytes
                | (1u << 20)        // pad_enable
                | (7u << 22)        // pad_interval: 256 DWORDs (one row)
                | (11u << 25));     // pad_amount: 12 DWORDs
    g1[1] = (int)(256u << 16);      // tensor_dim0 = 256 elements
    g1[2] = (int)(1024u << 16);     // tensor_dim1 = 1024 rows
    g1[3] = (int)(256u << 16);      // tile_dim0 = 256
    g1[4] = 64;                     // tile_dim1 = 64, tile_dim2 = 0
    g1[5] = 256;                    // tensor_dim0_stride = 256 elements
    g1[6] = 0;                      // tensor_dim1_stride unused (2D)
    g1[7] = 0;
    v4i gz4 = (v4i){0, 0, 0, 0};            // groups 2/3: unused (2D tensor)
    v8i gz8 = (v8i){0, 0, 0, 0, 0, 0, 0, 0};
    // 6-arg variant (clang-23 / therock-10.0): (g0, g1, g2, g3, g?, cpol)
    __builtin_amdgcn_tensor_load_to_lds(g0, g1, gz4, gz4, gz8, 0);
}
#endif

__global__ __launch_bounds__(256)
void frechet_mean_kernel(const float* __restrict__ points,
                         const float* __restrict__ weights,
                         float* __restrict__ out) {
    extern __shared__ float smem[];
    float* tile0 = smem;                          // TILE_N * STRIDE
    float* tile1 = tile0 + TILE_N * STRIDE;       // TILE_N * STRIDE
    float* wnorm = tile1 + TILE_N * STRIDE;       // N_
    float* meanv = wnorm + N_;                    // D_
    float* gradv = meanv + D_;                    // D_
    float* cos0  = gradv + D_;                    // TILE_N (D-half 0 partial)
    float* cos1  = cos0  + TILE_N;                // TILE_N (D-half 1 partial)
    float* a_arr = cos1  + TILE_N;                // TILE_N
    float* scal  = a_arr + TILE_N;                // scalars

    const int t       = threadIdx.x;
    const int lane    = t & 31;
    const int wv      = t >> 5;
    const int lane_lo = lane & 15;
    const int half    = lane >> 4;                // 0: lanes 0-15, 1: lanes 16-31
    const int b       = blockIdx.x;

    const float* P = points  + (size_t)b * N_ * D_;
    const float* W = weights + (size_t)b * N_;

#if USE_TDM
    const unsigned lds_base = (unsigned)__builtin_amdgcn_groupstaticsize();
    const unsigned off0 = lds_base;                                    // tile0
    const unsigned off1 = lds_base + (unsigned)(TILE_N * STRIDE * 4);  // tile1
#endif

    // ---------------- normalize weights ----------------
    if (t == 0) scal[0] = 0.f;
    __syncthreads();
    float part = 0.f;
    for (int i = t; i < N_; i += 256) { float w = W[i]; wnorm[i] = w; part += w; }
    atomicAdd(&scal[0], part);
    __syncthreads();
    const float inv_wsum = 1.0f / scal[0];
    for (int i = t; i < N_; i += 256) wnorm[i] *= inv_wsum;
    __syncthreads();

    // ---------------- Euclidean init, projected (thread t owns dim d = t) ---
    float m = 0.f;
    for (int n = 0; n < N_; ++n) m += wnorm[n] * P[(size_t)n * D_ + t];
    if (t == 0) scal[1] = 0.f;
    __syncthreads();
    atomicAdd(&scal[1], m * m);
    __syncthreads();
    meanv[t] = m / fmaxf(sqrtf(scal[1]), EPSX);
    __syncthreads();

    // ---------------- Riemannian GD iterations ----------------
    const int g   = wv >> 1;          // cos: row-group (16 rows) of the tile
    const int h   = wv & 1;           // cos: D-half handled by this wave
    const int d0A = wv * 32;          // grad: first 16-wide d column group
    const int d0B = wv * 32 + 16;     // grad: second 16-wide d column group

    for (int it = 0; it < ITERS; ++it) {
        v8f gaccA = {0,0,0,0,0,0,0,0};
        v8f gaccB = {0,0,0,0,0,0,0,0};
        float s_priv = 0.f;
        if (t == 0) { scal[1] = 0.f; scal[2] = 0.f; }

#if USE_TDM
        if (wv == 0) tdm_load_tile(P, off0);      // prologue: tile 0 in flight
#endif

        for (int ti = 0; ti < NTILES; ++ti) {
            const float* buf = (ti & 1) ? tile1 : tile0;
            __syncthreads();          // prior readers of the other buffer done
#if USE_TDM
            if (wv == 0) {
                if (ti + 1 < NTILES) {
                    // issue next tile behind current one, then wait: in-order
                    // TDM completion => count<=1 proves tile ti has landed.
                    tdm_load_tile(P + (size_t)(ti + 1) * TILE_N * D_,
                                  ((ti + 1) & 1) ? off1 : off0);
                    __builtin_amdgcn_s_wait_tensorcnt(1);
                } else {
                    __builtin_amdgcn_s_wait_tensorcnt(0);
                }
            }
#else
            // fallback: synchronous staging (coalesced float4 global -> LDS)
            {
                const float4* src = (const float4*)(P + (size_t)ti * TILE_N * D_);
                float* dbuf = (ti & 1) ? tile1 : tile0;
                #pragma unroll
                for (int p = 0; p < (TILE_N * D_ / 4) / 256; ++p) {
                    int lin = p * 256 + t;
                    int row = lin >> 6;
                    int c4  = lin & 63;
                    float4 v = src[row * 64 + c4];
                    float* dst = &dbuf[row * STRIDE + c4 * 4];
                    dst[0] = v.x; dst[1] = v.y; dst[2] = v.z; dst[3] = v.w;
                }
            }
#endif
            __syncthreads();          // publish tile ti to all waves

            // ---- cos_n = <P_n, mean> via WMMA, wave = (row-group g, D-half h)
            v8f cacc = {0,0,0,0,0,0,0,0};
            const float* arow = &buf[(g * 16 + lane_lo) * STRIDE + 2 * half];
            const float* brow = &meanv[2 * half];
            for (int kk = h * 128; kk < h * 128 + 128; kk += 4) {
                v2f a2 = *(const v2f*)(arow + kk);   // A: 16 rows x K=4 of P
                v2f b2 = *(const v2f*)(brow + kk);   // B: mean chunk, all cols
                cacc = __builtin_amdgcn_wmma_f32_16x16x4_f32(
                    false, a2, false, b2, (short)0, cacc, false, false);
            }
            if (lane_lo == 0) {                      // column 0 carries the dot
                float* dst = h ? cos1 : cos0;
                #pragma unroll
                for (int r = 0; r < 8; ++r) dst[g * 16 + half * 8 + r] = cacc[r];
            }
            __syncthreads();

            // ---- a_n = w_n * theta / ||u|| ; accumulate s = sum a_n cos_n ---
            if (t < TILE_N) {
                float c = cos0[t] + cos1[t];
                c = fminf(fmaxf(c, -1.0f + EPSX), 1.0f - EPSX);
                float theta = acosf(c);
                float un = fmaxf(sqrtf(fmaxf(1.0f - c * c, 0.0f)), EPSX);
                float a = wnorm[ti * TILE_N + t] * theta / un;
                a_arr[t] = a;
                s_priv += a * c;
            }
            __syncthreads();

            // ---- grad partial: sum_n a_n * P[n, d] via WMMA (persistent acc)
            for (int n0 = 0; n0 < TILE_N; n0 += 4) {
                v2f a2 = *(const v2f*)&a_arr[n0 + 2 * half]; // A rows replicated
                const float* r0 = &buf[(n0 + 2 * half) * STRIDE];
                const float* r1 = r0 + STRIDE;
                v2f bA; bA.x = r0[d0A + lane_lo]; bA.y = r1[d0A + lane_lo];
                v2f bB; bB.x = r0[d0B + lane_lo]; bB.y = r1[d0B + lane_lo];
                gaccA = __builtin_amdgcn_wmma_f32_16x16x4_f32(
                    false, a2, false, bA, (short)0, gaccA, false, false);
                gaccB = __builtin_amdgcn_wmma_f32_16x16x4_f32(
                    false, a2, false, bB, (short)0, gaccB, false, false);
            }
        }

        // ---- extract grad (row 0 of each accumulator) + reduce s ----
        __syncthreads();
        if (half == 0) {
            gradv[d0A + lane_lo] = gaccA[0];
            gradv[d0B + lane_lo] = gaccB[0];
        }
        if (t < TILE_N) atomicAdd(&scal[1], s_priv);
        __syncthreads();

        // ---- exp map: mean <- cos(|v|) mean + sin(|v|)/|v| * v ----
        float s  = scal[1];
        float v  = LR * (gradv[t] - s * meanv[t]);
        atomicAdd(&scal[2], v * v);
        __syncthreads();
        float tn = sqrtf(scal[2]);
        float mnew = cosf(tn) * meanv[t] + (sinf(tn) / fmaxf(tn, EPSX)) * v;
        __syncthreads();
        meanv[t] = mnew;
        __syncthreads();
    }

    out[(size_t)b * D_ + t] = meanv[t];
}

extern "C" void kernel_launch(void* const* d_in, const int* in_sizes, int n_in,
                              void* d_out, int out_size, void* d_ws, size_t ws_size,
                              hipStream_t stream) {
    const float* points  = (const float*)d_in[0];   // (B, N, D) fp32
    const float* weights = (const float*)d_in[1];   // (B, N)    fp32
    float* out = (float*)d_out;                     // (B, D)    fp32
    (void)in_sizes; (void)n_in; (void)out_size; (void)d_ws; (void)ws_size;

    const size_t shmem =
        (size_t)(2 * TILE_N * STRIDE + N_ + D_ + D_ + 3 * TILE_N + 8) * sizeof(float);
    frechet_mean_kernel<<<B_, 256, shmem, stream>>>(points, weights, out);
}